// ConvNetLayer_30769145708767
// MI455X (gfx1250) — compile-verified
//
#include <hip/hip_runtime.h>

#define H     128
#define NAT   300000
#define NBOND 600000
#define NBRS  6
#define EPSV  1e-5f
#define WROW  132            // padded LDS row stride (floats) -> conflict-free b64 reads
#define WSZ   (H * WROW)     // one 128x128 weight matrix in LDS (floats)

typedef float v2f __attribute__((ext_vector_type(2)));
typedef float v8f __attribute__((ext_vector_type(8)));

__device__ __forceinline__ v8f wmma4(v2f a, v2f b, v8f c) {
    // D = A(16x4) x B(4x16) + C, full fp32
    return __builtin_amdgcn_wmma_f32_16x16x4_f32(false, a, false, b, (short)0, c,
                                                 false, false);
}

__device__ __forceinline__ float sigf(float x) {
    return 1.f / (1.f + __expf(-x));
}

// ---------------------------------------------------------------- stats zero
__global__ void zero_stats_k(float* stats) {
    int t = threadIdx.x;
    if (t < 512) stats[t] = 0.f;   // bond_sum/sumsq, atom_sum/sumsq
}

// ---------------------------------------------------------------- bond GEMM
// edge_syn = bond@A^T + atom[i0]@B^T + atom[i1]@C^T + (Ab+Bb+Cb)
// Each wave computes TWO 16-row strips (M=32) so every B-fragment ds_load
// feeds two WMMAs.
__global__ void __launch_bounds__(256, 1)
bond_gemm_k(const float* __restrict__ atom_in,
            const float* __restrict__ bond_in,
            const int*   __restrict__ badj,   // (NBOND,2)
            const float* __restrict__ Aw, const float* __restrict__ Ab,
            const float* __restrict__ Bw, const float* __restrict__ Bb,
            const float* __restrict__ Cw, const float* __restrict__ Cb,
            float* __restrict__ edge_syn,
            float* __restrict__ bsum, float* __restrict__ bsumsq) {
    extern __shared__ float smem[];
    const int tid = threadIdx.x;
    for (int i = tid; i < H * H; i += blockDim.x) {
        int r = i >> 7, c = i & 127;
        smem[0 * WSZ + r * WROW + c] = Aw[i];
        smem[1 * WSZ + r * WROW + c] = Bw[i];
        smem[2 * WSZ + r * WROW + c] = Cw[i];
    }
    __syncthreads();

    const int lane = tid & 31;
    const int wave = tid >> 5;
    const int pair = blockIdx.x * (blockDim.x >> 5) + wave;  // 32-row pair
    if (pair >= NBOND / 32) return;        // wave-uniform: EXEC stays all-ones

    const int m    = lane & 15;
    const int hf   = lane >> 4;            // 0 or 1
    const int koff = hf * 2;               // A/B frag K sub-offset {0,2}
    const int row0 = pair * 32 + m;
    const int row1 = row0 + 16;

    const float* pb0 = bond_in + (size_t)row0 * H + koff;
    const float* pb1 = bond_in + (size_t)row1 * H + koff;
    const int i00 = badj[2 * row0 + 0];
    const int i01 = badj[2 * row0 + 1];
    const int i10 = badj[2 * row1 + 0];
    const int i11 = badj[2 * row1 + 1];
    const float* px0 = atom_in + (size_t)i00 * H + koff;
    const float* py0 = atom_in + (size_t)i01 * H + koff;
    const float* px1 = atom_in + (size_t)i10 * H + koff;
    const float* py1 = atom_in + (size_t)i11 * H + koff;

    v8f acc0[8] = {};
    v8f acc1[8] = {};
    const float* wb = smem + m * WROW + koff;

    for (int k = 0; k < H; k += 4) {
        v2f a00 = *(const v2f*)(pb0 + k);
        v2f a01 = *(const v2f*)(px0 + k);
        v2f a02 = *(const v2f*)(py0 + k);
        v2f a10 = *(const v2f*)(pb1 + k);
        v2f a11 = *(const v2f*)(px1 + k);
        v2f a12 = *(const v2f*)(py1 + k);
#pragma unroll
        for (int nt = 0; nt < 8; ++nt) {
            const float* w = wb + nt * 16 * WROW + k;
            v2f bA = *(const v2f*)(w + 0 * WSZ);
            v2f bB = *(const v2f*)(w + 1 * WSZ);
            v2f bC = *(const v2f*)(w + 2 * WSZ);
            acc0[nt] = wmma4(a00, bA, acc0[nt]);
            acc1[nt] = wmma4(a10, bA, acc1[nt]);
            acc0[nt] = wmma4(a01, bB, acc0[nt]);
            acc1[nt] = wmma4(a11, bB, acc1[nt]);
            acc0[nt] = wmma4(a02, bC, acc0[nt]);
            acc1[nt] = wmma4(a12, bC, acc1[nt]);
        }
    }

    // C/D layout: lanes>=16 hold M=8..15 of each 16-row tile
    const int rb0 = pair * 32 + hf * 8;
    const int rb1 = rb0 + 16;
#pragma unroll
    for (int nt = 0; nt < 8; ++nt) {
        const int o = nt * 16 + m;
        const float bias = Ab[o] + Bb[o] + Cb[o];
        float s = 0.f, s2 = 0.f;
#pragma unroll
        for (int v = 0; v < 8; ++v) {
            float d0 = acc0[nt][v] + bias;
            float d1 = acc1[nt][v] + bias;
            edge_syn[(size_t)(rb0 + v) * H + o] = d0;
            edge_syn[(size_t)(rb1 + v) * H + o] = d1;
            s += d0 + d1;
            s2 += d0 * d0 + d1 * d1;
        }
        s  += __shfl_xor(s, 16, 32);
        s2 += __shfl_xor(s2, 16, 32);
        if (hf == 0) {
            atomicAdd(&bsum[o],   s);
            atomicAdd(&bsumsq[o], s2);
        }
    }
}

// ---------------------------------------------------------------- atom GEMMs
// ux_out (=d_out atom region) = atom@U^T + Ub ; vx = atom@V^T + Vb
// A-fragment shared across the U and V chains (ds:wmma = 1:1 already).
__global__ void __launch_bounds__(256, 1)
atom_gemm_k(const float* __restrict__ atom_in,
            const float* __restrict__ Uw, const float* __restrict__ Ub,
            const float* __restrict__ Vw, const float* __restrict__ Vb,
            float* __restrict__ ux_out,
            float* __restrict__ vx) {
    extern __shared__ float smem[];
    const int tid = threadIdx.x;
    for (int i = tid; i < H * H; i += blockDim.x) {
        int r = i >> 7, c = i & 127;
        smem[0 * WSZ + r * WROW + c] = Uw[i];
        smem[1 * WSZ + r * WROW + c] = Vw[i];
    }
    __syncthreads();

    const int lane  = tid & 31;
    const int wave  = tid >> 5;
    const int strip = blockIdx.x * (blockDim.x >> 5) + wave;
    if (strip >= NAT / 16) return;

    const int m    = lane & 15;
    const int hf   = lane >> 4;
    const int koff = hf * 2;
    const int row  = strip * 16 + m;

    const float* pa = atom_in + (size_t)row * H + koff;

    v8f accU[8] = {};
    v8f accV[8] = {};
    const float* wb = smem + m * WROW + koff;

    for (int k = 0; k < H; k += 4) {
        v2f a = *(const v2f*)(pa + k);
#pragma unroll
        for (int nt = 0; nt < 8; ++nt) {
            const float* w = wb + nt * 16 * WROW + k;
            v2f bU = *(const v2f*)(w + 0 * WSZ);
            v2f bV = *(const v2f*)(w + 1 * WSZ);
            accU[nt] = wmma4(a, bU, accU[nt]);
            accV[nt] = wmma4(a, bV, accV[nt]);
        }
    }

    const int rbase = strip * 16 + hf * 8;
#pragma unroll
    for (int nt = 0; nt < 8; ++nt) {
        const int o = nt * 16 + m;
        const float bu = Ub[o];
        const float bv = Vb[o];
#pragma unroll
        for (int v = 0; v < 8; ++v) {
            size_t idx = (size_t)(rbase + v) * H + o;
            ux_out[idx] = accU[nt][v] + bu;
            vx[idx]     = accV[nt][v] + bv;
        }
    }
}

// ---------------------------------------------------------------- atom message
// atom_syn(in d_out) += sum_k sigmoid(edge_syn[abadj]) * vx[aadj]; BN stats
__global__ void atom_msg_k(const float* __restrict__ edge_syn,
                           const float* __restrict__ vx,
                           const int*   __restrict__ aadj,
                           const int*   __restrict__ abadj,
                           float* __restrict__ atom_syn,
                           float* __restrict__ asum, float* __restrict__ asumsq) {
    __shared__ float lsum[H];
    __shared__ float lsq[H];
    const int tid = threadIdx.x;
    if (tid < H) { lsum[tid] = 0.f; lsq[tid] = 0.f; }
    __syncthreads();

    const int lane = tid & 31;
    const int wave = tid >> 5;
    const int wglob = blockIdx.x * (blockDim.x >> 5) + wave;
    const int nwave = (gridDim.x * blockDim.x) >> 5;
    const int fo = lane * 4;  // this lane's 4 features

    float4 s  = make_float4(0.f, 0.f, 0.f, 0.f);
    float4 s2 = make_float4(0.f, 0.f, 0.f, 0.f);

    for (int n = wglob; n < NAT; n += nwave) {
        const int* pa = aadj  + n * NBRS;
        const int* pb = abadj + n * NBRS;
        float4 acc = *(const float4*)(atom_syn + (size_t)n * H + fo);
#pragma unroll
        for (int k = 0; k < NBRS; ++k) {
            const int bi = pb[k];
            const int ai = pa[k];
            float4 g = *(const float4*)(edge_syn + (size_t)bi * H + fo);
            float4 w = *(const float4*)(vx       + (size_t)ai * H + fo);
            acc.x += sigf(g.x) * w.x;
            acc.y += sigf(g.y) * w.y;
            acc.z += sigf(g.z) * w.z;
            acc.w += sigf(g.w) * w.w;
        }
        *(float4*)(atom_syn + (size_t)n * H + fo) = acc;
        s.x += acc.x; s.y += acc.y; s.z += acc.z; s.w += acc.w;
        s2.x += acc.x * acc.x; s2.y += acc.y * acc.y;
        s2.z += acc.z * acc.z; s2.w += acc.w * acc.w;
    }

    atomicAdd(&lsum[fo + 0], s.x);  atomicAdd(&lsq[fo + 0], s2.x);
    atomicAdd(&lsum[fo + 1], s.y);  atomicAdd(&lsq[fo + 1], s2.y);
    atomicAdd(&lsum[fo + 2], s.z);  atomicAdd(&lsq[fo + 2], s2.z);
    atomicAdd(&lsum[fo + 3], s.w);  atomicAdd(&lsq[fo + 3], s2.w);
    __syncthreads();
    if (tid < H) {
        atomicAdd(&asum[tid],   lsum[tid]);
        atomicAdd(&asumsq[tid], lsq[tid]);
    }
}

// ---------------------------------------------------------------- BN finalize
__global__ void finalize_k(const float* __restrict__ sum, const float* __restrict__ sumsq,
                           const float* __restrict__ gamma, const float* __restrict__ beta,
                           float invN, float* __restrict__ scale, float* __restrict__ shift) {
    int t = threadIdx.x;
    if (t < H) {
        float mu  = sum[t] * invN;
        float var = sumsq[t] * invN - mu * mu;
        float rs  = rsqrtf(var + EPSV);
        float sc  = gamma[t] * rs;
        scale[t] = sc;
        shift[t] = beta[t] - mu * sc;
    }
}

// ---------------------------------------------------------------- outputs
__global__ void atom_out_k(const float* __restrict__ atom_in,
                           const float* __restrict__ scale, const float* __restrict__ shift,
                           float* __restrict__ out) {   // in-place: atom_syn -> output
    const size_t n4 = (size_t)NAT * H / 4;
    size_t stride = (size_t)gridDim.x * blockDim.x;
    for (size_t i = (size_t)blockIdx.x * blockDim.x + threadIdx.x; i < n4; i += stride) {
        const int ob = (int)(i & 31);
        float4 sc = ((const float4*)scale)[ob];
        float4 sh = ((const float4*)shift)[ob];
        float4 x  = ((const float4*)out)[i];
        float4 a  = ((const float4*)atom_in)[i];
        float4 r;
        r.x = fmaxf(x.x * sc.x + sh.x, 0.f) + a.x;
        r.y = fmaxf(x.y * sc.y + sh.y, 0.f) + a.y;
        r.z = fmaxf(x.z * sc.z + sh.z, 0.f) + a.z;
        r.w = fmaxf(x.w * sc.w + sh.w, 0.f) + a.w;
        ((float4*)out)[i] = r;
    }
}

__global__ void bond_out_k(const float* __restrict__ edge_syn,
                           const float* __restrict__ scale, const float* __restrict__ shift,
                           float* __restrict__ out) {
    const size_t n4 = (size_t)NBOND * H / 4;
    size_t stride = (size_t)gridDim.x * blockDim.x;
    for (size_t i = (size_t)blockIdx.x * blockDim.x + threadIdx.x; i < n4; i += stride) {
        const int ob = (int)(i & 31);
        float4 sc = ((const float4*)scale)[ob];
        float4 sh = ((const float4*)shift)[ob];
        float4 x  = ((const float4*)edge_syn)[i];
        float4 r;
        r.x = fmaxf(x.x * sc.x + sh.x, 0.f);
        r.y = fmaxf(x.y * sc.y + sh.y, 0.f);
        r.z = fmaxf(x.z * sc.z + sh.z, 0.f);
        r.w = fmaxf(x.w * sc.w + sh.w, 0.f);
        ((float4*)out)[i] = r;
    }
}

// ---------------------------------------------------------------- launch
extern "C" void kernel_launch(void* const* d_in, const int* in_sizes, int n_in,
                              void* d_out, int out_size, void* d_ws, size_t ws_size,
                              hipStream_t stream) {
    const float* atom_in = (const float*)d_in[0];
    const float* bond_in = (const float*)d_in[1];
    const int*   aadj    = (const int*)d_in[2];
    const int*   abadj   = (const int*)d_in[3];
    const int*   badj    = (const int*)d_in[4];
    const float* Uw = (const float*)d_in[5];  const float* Ub = (const float*)d_in[6];
    const float* Vw = (const float*)d_in[7];  const float* Vb = (const float*)d_in[8];
    const float* Aw = (const float*)d_in[9];  const float* Ab = (const float*)d_in[10];
    const float* Bw = (const float*)d_in[11]; const float* Bb = (const float*)d_in[12];
    const float* Cw = (const float*)d_in[13]; const float* Cb = (const float*)d_in[14];
    const float* bgam = (const float*)d_in[15]; const float* bbet = (const float*)d_in[16];
    const float* agam = (const float*)d_in[17]; const float* abet = (const float*)d_in[18];

    float* out_atom = (float*)d_out;
    float* out_bond = out_atom + (size_t)NAT * H;

    // workspace: edge_syn | vx | stats(1024 floats)
    float* edge_syn = (float*)d_ws;
    float* vx       = edge_syn + (size_t)NBOND * H;
    float* stats    = vx + (size_t)NAT * H;
    float* bsum = stats;        float* bsumsq = stats + 128;
    float* asum = stats + 256;  float* asumsq = stats + 384;
    float* bscale = stats + 512; float* bshift = stats + 640;
    float* ascale = stats + 768; float* ashift = stats + 896;

    zero_stats_k<<<1, 512, 0, stream>>>(stats);

    // bond GEMM: 600000/32 = 18750 row-pairs, 8 waves/block
    bond_gemm_k<<<(NBOND / 32 + 7) / 8, 256, 3 * WSZ * sizeof(float), stream>>>(
        atom_in, bond_in, badj, Aw, Ab, Bw, Bb, Cw, Cb, edge_syn, bsum, bsumsq);

    finalize_k<<<1, 128, 0, stream>>>(bsum, bsumsq, bgam, bbet, 1.f / NBOND, bscale, bshift);

    // atom GEMMs: 300000/16 = 18750 strips
    atom_gemm_k<<<(NAT / 16 + 7) / 8, 256, 2 * WSZ * sizeof(float), stream>>>(
        atom_in, Uw, Ub, Vw, Vb, out_atom, vx);

    atom_msg_k<<<512, 256, 0, stream>>>(edge_syn, vx, aadj, abadj, out_atom, asum, asumsq);

    finalize_k<<<1, 128, 0, stream>>>(asum, asumsq, agam, abet, 1.f / NAT, ascale, ashift);

    atom_out_k<<<2048, 256, 0, stream>>>(atom_in, ascale, ashift, out_atom);
    bond_out_k<<<4096, 256, 0, stream>>>(edge_syn, bscale, bshift, out_bond);
}